// NormalLSTM_44633300140445
// MI455X (gfx1250) — compile-verified
//
#include <hip/hip_runtime.h>

#define PRED_LEN 12
#define HIDDEN 64
#define T_ENC 8
#define IN_DIM 2
#define WAVES_PER_BLOCK 8
#define ROWS_PER_WAVE 16
#define ROWS_PER_BLOCK (WAVES_PER_BLOCK * ROWS_PER_WAVE)   // 128
#define HROW_STRIDE 68   // 64 + 4 floats pad -> conflict-free A-layout reads

typedef __attribute__((ext_vector_type(2))) float v2f;
typedef __attribute__((ext_vector_type(4))) float v4f;
typedef __attribute__((ext_vector_type(8))) float v8f;

__device__ __forceinline__ float fast_rcp(float x) {
    return __builtin_amdgcn_rcpf(x);            // v_rcp_f32, ~1 ulp
}
__device__ __forceinline__ float sigm(float x) {
    return fast_rcp(1.0f + __expf(-x));
}
__device__ __forceinline__ float tanh_fast(float x) {
    return 1.0f - 2.0f * fast_rcp(__expf(2.0f * x) + 1.0f);
}

__device__ __forceinline__ v8f wmma_f32(v2f a, v2f b, v8f c) {
    // D = A(16x4 f32) * B(4x16 f32) + C(16x16 f32)
    return __builtin_amdgcn_wmma_f32_16x16x4_f32(
        /*neg_a=*/false, a, /*neg_b=*/false, b,
        /*c_mod=*/(short)0, c, /*reuse_a=*/false, /*reuse_b=*/false);
}

__launch_bounds__(256, 1)
__global__ void lstm_seq2seq_kernel(const float* __restrict__ obs,
                                    const float* __restrict__ W_ih,
                                    const float* __restrict__ W_hh,
                                    const float* __restrict__ b_ih,
                                    const float* __restrict__ b_hh,
                                    const float* __restrict__ fc_W,
                                    const float* __restrict__ fc_b,
                                    float* __restrict__ out)
{
    // W_hh pre-swizzled into B-operand layout:
    //   entry (t,k,lane) = { B[4k+2*(lane/16)+0][16t+lane%16], B[...+1][...] }
    //   where B[kk][n] = W_hh[n][kk]   (z = h @ W_hh^T)
    __shared__ v2f   sBw[16 * 16 * 32];                                // 64 KB
    // x-projection B operand: rows = {W_ih[:,0], W_ih[:,1], b_ih+b_hh, 0}
    __shared__ v2f   sBx[16 * 32];                                     // 4 KB
    __shared__ float sH[WAVES_PER_BLOCK * ROWS_PER_WAVE * HROW_STRIDE];// 34 KB
    __shared__ float sX[WAVES_PER_BLOCK * ROWS_PER_WAVE * 2];          // 1 KB
    __shared__ float sFcW[2 * HIDDEN];
    __shared__ float sFcB[2];

    const int tid = threadIdx.x;

    // ---- one-time weight preprocessing (per block) ----
    for (int idx = tid; idx < 16 * 16 * 32; idx += 256) {
        int t = idx >> 9;
        int k = (idx >> 5) & 15;
        int l = idx & 31;
        int n  = (t << 4) | (l & 15);
        int kk = (k << 2) + ((l >> 4) << 1);
        v2f w;
        w.x = W_hh[n * HIDDEN + kk];
        w.y = W_hh[n * HIDDEN + kk + 1];
        sBw[idx] = w;
    }
    for (int idx = tid; idx < 16 * 32; idx += 256) {
        int t = idx >> 5;
        int l = idx & 31;
        int n = (t << 4) | (l & 15);
        v2f w;
        if (l < 16) { w.x = W_ih[n * IN_DIM + 0]; w.y = W_ih[n * IN_DIM + 1]; }
        else        { w.x = b_ih[n] + b_hh[n];    w.y = 0.0f; }
        sBx[idx] = w;
    }
    for (int idx = tid; idx < 2 * HIDDEN; idx += 256) sFcW[idx] = fc_W[idx];
    if (tid < 2) sFcB[tid] = fc_b[tid];
    for (int idx = tid; idx < WAVES_PER_BLOCK * ROWS_PER_WAVE * HROW_STRIDE; idx += 256)
        sH[idx] = 0.0f;   // h0 = 0
    __syncthreads();

    const int wave      = tid >> 5;
    const int lane      = tid & 31;
    const int rowInWave = lane & 15;
    const int laneHalf  = lane >> 4;
    const int b0        = blockIdx.x * ROWS_PER_BLOCK + wave * ROWS_PER_WAVE;
    float* hbase = &sH[(wave * ROWS_PER_WAVE) * HROW_STRIDE];

    v8f cst[4];   // cell state, C-layout, cols 0..63
    #pragma unroll
    for (int j = 0; j < 4; ++j) cst[j] = (v8f){};

    // ---- one LSTM cell ----
    // Tiles processed in two groups of 8 (acc = 64 VGPRs live, stays <256 total).
    // Group (jp) covers gate-column tiles {4g + jp + dj : g=0..3, dj=0..1}, so each
    // group has matching i/f/g/o quadruples and can gate immediately. New h is
    // staged in registers; LDS h-buffer still holds OLD h for the 2nd group's
    // A-operand reads, then is published once at the end of the step.
    auto cell = [&](v2f ax) {
        float hnew[4][8];
        #pragma unroll
        for (int jp = 0; jp < 4; jp += 2) {
            v8f acc[8];
            v8f zero = {};
            #pragma unroll
            for (int dj = 0; dj < 2; ++dj) {
                #pragma unroll
                for (int g = 0; g < 4; ++g) {
                    int t = 4 * g + jp + dj;
                    v2f bx = sBx[(t << 5) | lane];
                    acc[dj * 4 + g] = wmma_f32(ax, bx, zero);  // x0*Wih0+x1*Wih1+bias
                }
            }
            #pragma unroll
            for (int k = 0; k < 16; ++k) {
                int kk = (k << 2) | (laneHalf << 1);
                v2f ah = *(const v2f*)&hbase[rowInWave * HROW_STRIDE + kk];
                #pragma unroll
                for (int dj = 0; dj < 2; ++dj) {
                    #pragma unroll
                    for (int g = 0; g < 4; ++g) {
                        int t = 4 * g + jp + dj;
                        v2f bw = sBw[(((t << 4) | k) << 5) | lane];
                        acc[dj * 4 + g] = wmma_f32(ah, bw, acc[dj * 4 + g]);
                    }
                }
            }
            // gate combine: acc[dj*4 + {0,1,2,3}] = i/f/g/o for column tile jp+dj
            #pragma unroll
            for (int dj = 0; dj < 2; ++dj) {
                int j = jp + dj;
                #pragma unroll
                for (int v = 0; v < 8; ++v) {
                    float ig = sigm(acc[dj * 4 + 0][v]);
                    float fg = sigm(acc[dj * 4 + 1][v]);
                    float gg = tanh_fast(acc[dj * 4 + 2][v]);
                    float og = sigm(acc[dj * 4 + 3][v]);
                    float cc = fg * cst[j][v] + ig * gg;
                    cst[j][v] = cc;
                    hnew[j][v] = og * tanh_fast(cc);
                }
            }
        }
        // all old-h A-operand reads complete -> publish new h (C-layout -> row-major)
        #pragma unroll
        for (int j = 0; j < 4; ++j) {
            #pragma unroll
            for (int v = 0; v < 8; ++v) {
                hbase[(v + (laneHalf << 3)) * HROW_STRIDE + (j << 4) + rowInWave] =
                    hnew[j][v];
            }
        }
        __syncthreads();
    };

    // ---- encoder: x_t = obs[:, t, :] ----
    v2f ax;
    #pragma unroll 1
    for (int s = 0; s < T_ENC; ++s) {
        v2f xv = *(const v2f*)&obs[(size_t)((b0 + rowInWave) * T_ENC + s) * IN_DIM];
        ax.x = laneHalf ? 1.0f : xv.x;   // lanes 16-31 carry the bias "1" column
        ax.y = laneHalf ? 0.0f : xv.y;
        cell(ax);
    }

    // ---- decoder: x0 = obs[:, -1, :] (already in ax), then feedback ----
    #pragma unroll 1
    for (int d = 0; d < PRED_LEN; ++d) {
        cell(ax);
        // pred = h @ fc_W^T + fc_b ; lane = (row, out) pair
        const float* hr = &hbase[rowInWave * HROW_STRIDE];
        const float* wr = &sFcW[laneHalf * HIDDEN];
        float p = sFcB[laneHalf];
        #pragma unroll
        for (int k = 0; k < HIDDEN; k += 4) {
            v4f h4 = *(const v4f*)&hr[k];
            v4f w4 = *(const v4f*)&wr[k];
            p += h4.x * w4.x + h4.y * w4.y + h4.z * w4.z + h4.w * w4.w;
        }
        out[(size_t)((b0 + rowInWave) * PRED_LEN + d) * 2 + laneHalf] = p;
        sX[((wave << 4) | rowInWave) * 2 + laneHalf] = p;
        __syncthreads();
        v2f xv = *(const v2f*)&sX[((wave << 4) | rowInWave) * 2];
        ax.x = laneHalf ? 1.0f : xv.x;
        ax.y = laneHalf ? 0.0f : xv.y;
    }
}

extern "C" void kernel_launch(void* const* d_in, const int* in_sizes, int n_in,
                              void* d_out, int out_size, void* d_ws, size_t ws_size,
                              hipStream_t stream) {
    const float* obs  = (const float*)d_in[0];
    const float* W_ih = (const float*)d_in[1];
    const float* W_hh = (const float*)d_in[2];
    const float* b_ih = (const float*)d_in[3];
    const float* b_hh = (const float*)d_in[4];
    const float* fc_W = (const float*)d_in[5];
    const float* fc_b = (const float*)d_in[6];
    float* outp = (float*)d_out;

    const int B = in_sizes[0] / (T_ENC * IN_DIM);   // 65536
    dim3 grid(B / ROWS_PER_BLOCK);                  // 512
    lstm_seq2seq_kernel<<<grid, 256, 0, stream>>>(obs, W_ih, W_hh, b_ih, b_hh,
                                                  fc_W, fc_b, outp);
}